// EntityCriterion_64098091925992
// MI455X (gfx1250) — compile-verified
//
#include <hip/hip_runtime.h>
#include <math.h>

#define B_ 64
#define Q_ 64
#define L_ 4096
#define E_ 32

typedef __attribute__((ext_vector_type(2))) float v2f;
typedef __attribute__((ext_vector_type(8))) float v8f;

// ---------------------------------------------------------------------------
// K1: fused argmax + logsumexp over each length-4096 row of start/end preds.
// One 256-thread block per row; each thread holds 16 floats in registers so
// the 128 MB of logits is read exactly once (HBM-roofline pass, ~5.5us).
// ---------------------------------------------------------------------------
__global__ void row_stats_kernel(const float* __restrict__ sp,
                                 const float* __restrict__ ep,
                                 int* __restrict__ sArg, int* __restrict__ eArg,
                                 float* __restrict__ sLse, float* __restrict__ eLse) {
  const int bid  = blockIdx.x;           // 0..8191
  const int head = bid >> 12;            // 0 = start, 1 = end
  const int row  = bid & 4095;
  const float* src = (head == 0 ? sp : ep) + (size_t)row * L_;
  const int t = threadIdx.x;             // 0..255

  float4 v[4];
#pragma unroll
  for (int c = 0; c < 4; ++c)
    v[c] = ((const float4*)src)[c * 256 + t];

  // thread-local argmax, first-occurrence semantics (strict >, ascending idx)
  float bv = -INFINITY; int bi = 0;
#pragma unroll
  for (int c = 0; c < 4; ++c) {
    int base = (c * 256 + t) * 4;
    float4 x = v[c];
    if (x.x > bv) { bv = x.x; bi = base;     }
    if (x.y > bv) { bv = x.y; bi = base + 1; }
    if (x.z > bv) { bv = x.z; bi = base + 2; }
    if (x.w > bv) { bv = x.w; bi = base + 3; }
  }

  __shared__ float rv[256];
  __shared__ int   ri[256];
  rv[t] = bv; ri[t] = bi;
  __syncthreads();
  for (int s = 128; s > 0; s >>= 1) {
    if (t < s) {
      float ov = rv[t + s]; int oi = ri[t + s];
      if (ov > rv[t] || (ov == rv[t] && oi < ri[t])) { rv[t] = ov; ri[t] = oi; }
    }
    __syncthreads();
  }
  const float M = rv[0];
  const int amax = ri[0];
  __syncthreads();

  float s = 0.f;
#pragma unroll
  for (int c = 0; c < 4; ++c) {
    float4 x = v[c];
    s += expf(x.x - M) + expf(x.y - M) + expf(x.z - M) + expf(x.w - M);
  }
  rv[t] = s;
  __syncthreads();
  for (int st = 128; st > 0; st >>= 1) {
    if (t < st) rv[t] += rv[t + st];
    __syncthreads();
  }
  if (t == 0) {
    float lse = M + logf(rv[0]);
    if (head == 0) { sArg[row] = amax; sLse[row] = lse; }
    else           { eArg[row] = amax; eLse[row] = lse; }
  }
}

// ---------------------------------------------------------------------------
// K2: tag softmax probs + logsumexp (tiny: 512 KB). One thread per row of 32.
// ---------------------------------------------------------------------------
__global__ void tag_stats_kernel(const float* __restrict__ tp,
                                 float* __restrict__ tLse,
                                 float* __restrict__ P) {
  int row = blockIdx.x * blockDim.x + threadIdx.x;
  if (row >= B_ * Q_) return;
  const float* x = tp + row * E_;
  float m = -INFINITY;
#pragma unroll
  for (int e = 0; e < E_; ++e) m = fmaxf(m, x[e]);
  float ex[E_];
  float s = 0.f;
#pragma unroll
  for (int e = 0; e < E_; ++e) { ex[e] = expf(x[e] - m); s += ex[e]; }
  float inv = 1.f / s;
#pragma unroll
  for (int e = 0; e < E_; ++e) P[row * E_ + e] = ex[e] * inv;
  tLse[row] = m + logf(s);
}

// ---------------------------------------------------------------------------
// K3: build the [64,64,64] cost matrix. Class cost computed as an exact
// one-hot GEMM on the matrix pipe:  C = P(64x32) x OneHot(32x64) using
// V_WMMA_F32_16X16X4_F32 (bit-exact: one nonzero product per output).
// One block per sample, 8 waves, each wave owns two 16x16 C tiles.
// ---------------------------------------------------------------------------
__global__ void cost_build_kernel(const int* __restrict__ sArg, const int* __restrict__ eArg,
                                  const int* __restrict__ sl, const int* __restrict__ el,
                                  const int* __restrict__ tl,
                                  const float* __restrict__ P,
                                  float* __restrict__ cost) {
  const int b    = blockIdx.x;
  const int t    = threadIdx.x;     // 0..255
  const int lane = t & 31;
  const int wave = t >> 5;          // 0..7

  __shared__ float Pl[Q_][E_];       // 64x32 softmax probs
  __shared__ float OneHot[E_][Q_];   // OneHot[k][j] = (tag_label[b][j]==k)
  __shared__ float ssA[Q_], seA[Q_], lsA[Q_], leA[Q_];

#pragma unroll
  for (int k = 0; k < 8; ++k) {
    int idx = t + k * 256;           // 0..2047
    ((float*)Pl)[idx]     = P[b * (Q_ * E_) + idx];
    ((float*)OneHot)[idx] = 0.f;
  }
  __syncthreads();
  if (t < Q_) {
    ssA[t] = (float)sArg[b * Q_ + t];
    seA[t] = (float)eArg[b * Q_ + t];
    lsA[t] = (float)sl[b * Q_ + t];
    leA[t] = (float)el[b * Q_ + t];
    OneHot[tl[b * Q_ + t]][t] = 1.f;
  }
  __syncthreads();

  const int mrow  = (lane & 15);         // row within tile (A) / col within tile (B)
  const int khalf = (lane >> 4) << 1;    // lanes 16-31 carry K+2,K+3 (ISA 16x4 f32 layout)
  const int half  = (lane >> 4) << 3;    // C/D: lanes 16-31 hold M+8

#pragma unroll
  for (int pass = 0; pass < 2; ++pass) {
    const int tile = wave + pass * 8;    // 0..15
    const int tm = tile >> 2, tn = tile & 3;
    const int m = tm * 16 + mrow;
    const int n = tn * 16 + mrow;
    v8f c = {};
#pragma unroll
    for (int kk = 0; kk < 8; ++kk) {     // K = 32 in steps of 4
      const int k0 = kk * 4 + khalf;
      v2f a;  a.x  = Pl[m][k0];        a.y  = Pl[m][k0 + 1];
      v2f bb; bb.x = OneHot[k0][n];    bb.y = OneHot[k0 + 1][n];
      c = __builtin_amdgcn_wmma_f32_16x16x4_f32(
            false, a, false, bb, (short)0, c, false, false);
    }
#pragma unroll
    for (int r = 0; r < 8; ++r) {
      const int i = tm * 16 + r + half;  // pred row
      const int j = n;                   // label col
      const float klass = -c[r];         // class cost = -softmax(tag)[i, tag_label[j]]
      const float ss = ssA[i], se = seA[i], ls = lsA[j], le = leA[j];
      const float span = fabsf(ss - ls) + fabsf(se - le);
      const float pl = fminf(ss, se), pr = fmaxf(ss, se);
      const float ll = fminf(ls, le), lr = fmaxf(ls, le);
      const float il = fmaxf(pl, ll);
      const float inter = fmaxf(pr - il, 0.f);          // bug-faithful: i_right = p_right
      const float ul = fminf(pl, ll), ur = fmaxf(pr, lr);
      const float uni = fmaxf(ur - ul, 1e-10f);
      cost[b * (Q_ * Q_) + i * Q_ + j] = span + (-inter / uni) + klass;
    }
  }
}

// ---------------------------------------------------------------------------
// K4: Hungarian (Jonker-style potentials), one block per sample (64 threads).
// Serial augmenting-path structure; column-parallel minv update / argmin /
// potential updates in LDS. Double precision to match the numpy solve.
// ---------------------------------------------------------------------------
__global__ void hungarian_kernel(const float* __restrict__ cost, int* __restrict__ jperm) {
  const int b = blockIdx.x;
  const int t = threadIdx.x;           // 0..63
  const double INFD = 1e18;

  __shared__ double cst[Q_][Q_];       // 32 KB
  __shared__ double u[Q_ + 1], v[Q_ + 1], minv[Q_ + 1];
  __shared__ int p[Q_ + 1], way[Q_ + 1];
  __shared__ unsigned char used[Q_ + 1];
  __shared__ double redV[Q_];
  __shared__ int redI[Q_];
  __shared__ int s_j0, s_i0;
  __shared__ double s_delta;

  for (int cc = 0; cc < Q_; ++cc)
    cst[t][cc] = (double)cost[b * (Q_ * Q_) + t * Q_ + cc];
  for (int jj = t; jj <= Q_; jj += Q_) { u[jj] = 0.0; v[jj] = 0.0; p[jj] = 0; way[jj] = 0; }
  __syncthreads();

  for (int i = 1; i <= Q_; ++i) {
    if (t == 0) { p[0] = i; s_j0 = 0; }
    for (int jj = t; jj <= Q_; jj += Q_) { minv[jj] = INFD; used[jj] = 0; }
    __syncthreads();
    while (true) {
      if (t == 0) { used[s_j0] = 1; s_i0 = p[s_j0]; }
      __syncthreads();
      const int i0 = s_i0;
      const int j0 = s_j0;
      const int j  = t + 1;
      const bool freej = (used[j] == 0);
      if (freej) {
        double cur = cst[i0 - 1][j - 1] - u[i0] - v[j];
        if (cur < minv[j]) { minv[j] = cur; way[j] = j0; }
      }
      __syncthreads();
      redV[t] = freej ? minv[j] : INFD;
      redI[t] = j;
      __syncthreads();
      for (int s = 32; s > 0; s >>= 1) {   // argmin, first-occurrence tie-break
        if (t < s) {
          if (redV[t + s] < redV[t] ||
              (redV[t + s] == redV[t] && redI[t + s] < redI[t])) {
            redV[t] = redV[t + s]; redI[t] = redI[t + s];
          }
        }
        __syncthreads();
      }
      if (t == 0) { s_j0 = redI[0]; s_delta = redV[0]; }
      __syncthreads();
      const double delta = s_delta;
      const int j1 = s_j0;
      for (int jj = t; jj <= Q_; jj += Q_) {
        if (used[jj]) { u[p[jj]] += delta; v[jj] -= delta; }  // p[] distinct: race-free
        else if (jj >= 1) minv[jj] -= delta;
      }
      __syncthreads();
      if (p[j1] == 0) break;               // uniform read
    }
    if (t == 0) {                          // augment along 'way'
      int j0 = s_j0;
      while (j0) { int j1 = way[j0]; p[j0] = p[j1]; j0 = j1; }
    }
    __syncthreads();
  }
  jperm[b * Q_ + (p[t + 1] - 1)] = t;      // col_of_row
}

// ---------------------------------------------------------------------------
// K5: cross-entropy via lse - logit[target]; scattered gathers hit L2
// (both 64MB logits tensors fit in the 192MB L2 after K1 streamed them).
// ---------------------------------------------------------------------------
__global__ void ce_kernel(const float* __restrict__ sp, const float* __restrict__ ep,
                          const float* __restrict__ tp,
                          const int* __restrict__ sl, const int* __restrict__ el,
                          const int* __restrict__ tl,
                          const float* __restrict__ sLse, const float* __restrict__ eLse,
                          const float* __restrict__ tLse,
                          const int* __restrict__ jperm,
                          float* __restrict__ perS, float* __restrict__ out) {
  const int b = blockIdx.x, q = threadIdx.x;
  const int idx = b * Q_ + q;
  const int jq = jperm[idx];
  const int ts = sl[b * Q_ + jq], te = el[b * Q_ + jq], tt = tl[b * Q_ + jq];
  float ce = (sLse[idx] - sp[(size_t)idx * L_ + ts])
           + (eLse[idx] - ep[(size_t)idx * L_ + te])
           + (tLse[idx] - tp[idx * E_ + tt]);
  __shared__ float red[Q_];
  red[q] = ce;
  __syncthreads();
  for (int s = 32; s > 0; s >>= 1) { if (q < s) red[q] += red[q + s]; __syncthreads(); }
  if (q == 0) perS[b] = red[0] / (float)Q_;
  out[1 + idx] = (float)jq;                 // permutation output
}

__global__ void final_kernel(const float* __restrict__ perS, float* __restrict__ out) {
  const int t = threadIdx.x;                // 64 threads
  __shared__ float red[B_];
  red[t] = perS[t];
  __syncthreads();
  for (int s = 32; s > 0; s >>= 1) { if (t < s) red[t] += red[t + s]; __syncthreads(); }
  if (t == 0) out[0] = red[0] / (float)B_;
}

// ---------------------------------------------------------------------------
extern "C" void kernel_launch(void* const* d_in, const int* in_sizes, int n_in,
                              void* d_out, int out_size, void* d_ws, size_t ws_size,
                              hipStream_t stream) {
  const float* sp = (const float*)d_in[0];   // [64,64,4096]
  const float* ep = (const float*)d_in[1];   // [64,64,4096]
  const float* tp = (const float*)d_in[2];   // [64,64,32]
  const int*   sl = (const int*)d_in[3];     // [64,64]
  const int*   el = (const int*)d_in[4];
  const int*   tl = (const int*)d_in[5];
  float* out = (float*)d_out;                // [0]=loss, [1..4096]=perm

  // workspace layout (~1.7 MB)
  int*   sArg = (int*)d_ws;                  // 4096
  int*   eArg = sArg + 4096;                 // 4096
  float* sLse = (float*)(eArg + 4096);       // 4096
  float* eLse = sLse + 4096;                 // 4096
  float* tLse = eLse + 4096;                 // 4096
  float* P    = tLse + 4096;                 // 4096*32
  float* cost = P + 4096 * 32;               // 64*64*64
  int*   jprm = (int*)(cost + 64 * 64 * 64); // 4096
  float* perS = (float*)(jprm + 4096);       // 64

  row_stats_kernel <<<2 * B_ * Q_, 256, 0, stream>>>(sp, ep, sArg, eArg, sLse, eLse);
  tag_stats_kernel <<<(B_ * Q_ + 255) / 256, 256, 0, stream>>>(tp, tLse, P);
  cost_build_kernel<<<B_, 256, 0, stream>>>(sArg, eArg, sl, el, tl, P, cost);
  hungarian_kernel <<<B_, Q_, 0, stream>>>(cost, jprm);
  ce_kernel        <<<B_, Q_, 0, stream>>>(sp, ep, tp, sl, el, tl, sLse, eLse, tLse, jprm, perS, out);
  final_kernel     <<<1, B_, 0, stream>>>(perS, out);
}